// MinimalHPM_28312424415585
// MI455X (gfx1250) — compile-verified
//
#include <hip/hip_runtime.h>
#include <hip/hip_bf16.h>

// ---------------------------------------------------------------------------
// Gaussian-tube ray readout:  out[b,c] = sum_n k(b,n) * memory[n,c]
//   k = exp(-r2/(2*0.5^2)) * exp(-max(t,0)/2)  ==  exp(-2*r2 - 0.5*max(t,0))
// with r2 = |p-o|^2 - t^2, t = (p-o).d   (d unit length).
// Grid coords come from the linear index (meshgrid 'ij'), no grid load needed.
//
// The k @ memory contraction runs on the matrix pipe via V_WMMA_F32_16X16X4_F32:
//   A (16x4 f32)  = k for 16 rays x 4 grid points
//   B (4x16 f32)  = memory rows for those 4 points (16 channels)
//   C (16x16 f32) = per-wave partial output tile (two tiles cover B=32 rays)
//
// Loop trip count is a compile-time constant and the induction variable is
// wave-uniform, so loop control is scalar (s_cmp/s_cbranch) and EXEC stays
// all-1s for every WMMA, as the ISA requires.
// ---------------------------------------------------------------------------

typedef __attribute__((ext_vector_type(2))) float v2f;
typedef __attribute__((ext_vector_type(8))) float v8f;

#define NPTS   (128 * 128 * 128)   // 2^21 grid points
#define NCH    16
#define NRAYS  32

#define BLOCKS        512
#define THREADS       256
#define TOTAL_WAVES   (BLOCKS * (THREADS / 32))   // 4096
#define STEP          (TOTAL_WAVES * 4)           // 16384 points consumed per sweep
#define ITERS         (NPTS / STEP)               // 128, identical for every wave

__device__ __forceinline__ float tube_k(float px, float py, float pz,
                                        float ox, float oy, float oz,
                                        float dx, float dy, float dz) {
    const float ux = px - ox;
    const float uy = py - oy;
    const float uz = pz - oz;
    const float t  = fmaf(ux, dx, fmaf(uy, dy, uz * dz));
    const float l2 = fmaf(ux, ux, fmaf(uy, uy, uz * uz));
    const float r2 = fmaf(-t, t, l2);                       // |u|^2 - t^2
    const float arg = fmaf(-2.0f, r2, -0.5f * fmaxf(t, 0.0f));
    return __expf(arg);                                      // v_exp_f32
}

__global__ void __launch_bounds__(THREADS)
hpm_zero_out(float* __restrict__ out) {
    const int i = blockIdx.x * blockDim.x + threadIdx.x;
    if (i < NRAYS * NCH) out[i] = 0.0f;
}

__global__ void __launch_bounds__(THREADS)
hpm_tube_read(const float* __restrict__ memory,
              const float* __restrict__ ray_o,
              const float* __restrict__ ray_d,
              float* __restrict__ out) {
    const int lane = threadIdx.x & 31;
    const int m    = lane & 15;       // ray-within-tile / channel index
    const int half = lane >> 4;       // selects K pair {0,1} vs {2,3}
    const int wave = blockIdx.x * (THREADS >> 5) + (threadIdx.x >> 5);

    // Ray data for ray m (tile 0) and ray m+16 (tile 1).
    const float ox0 = ray_o[m * 3 + 0],        oy0 = ray_o[m * 3 + 1],        oz0 = ray_o[m * 3 + 2];
    const float dx0 = ray_d[m * 3 + 0],        dy0 = ray_d[m * 3 + 1],        dz0 = ray_d[m * 3 + 2];
    const float ox1 = ray_o[(m + 16) * 3 + 0], oy1 = ray_o[(m + 16) * 3 + 1], oz1 = ray_o[(m + 16) * 3 + 2];
    const float dx1 = ray_d[(m + 16) * 3 + 0], dy1 = ray_d[(m + 16) * 3 + 1], dz1 = ray_d[(m + 16) * 3 + 2];

    v8f acc0 = {};   // rays 0..15  x channels 0..15
    v8f acc1 = {};   // rays 16..31 x channels 0..15

    const int lane_off = wave * 4 + (half << 1);   // this lane's first point in sweep 0

    #pragma unroll 2
    for (int it = 0; it < ITERS; ++it) {
        // This lane owns grid points pa and pa+1 (A layout: v0 -> K={0,2}, v1 -> K={1,3}).
        const int pa = lane_off + it * STEP;
        const int pb = pa + 1;

        const float px0 = (float)(pa >> 14);
        const float py0 = (float)((pa >> 7) & 127);
        const float pz0 = (float)(pa & 127);
        const float px1 = (float)(pb >> 14);
        const float py1 = (float)((pb >> 7) & 127);
        const float pz1 = (float)(pb & 127);

        v2f a0, a1, b;
        a0.x = tube_k(px0, py0, pz0, ox0, oy0, oz0, dx0, dy0, dz0);
        a0.y = tube_k(px1, py1, pz1, ox0, oy0, oz0, dx0, dy0, dz0);
        a1.x = tube_k(px0, py0, pz0, ox1, oy1, oz1, dx1, dy1, dz1);
        a1.y = tube_k(px1, py1, pz1, ox1, oy1, oz1, dx1, dy1, dz1);

        // B matrix: 4 memory rows x 16 channels (same K-pair split as A).
        b.x = memory[pa * NCH + m];
        b.y = memory[pb * NCH + m];

        // Pull the next sweep's tile toward the caches (global_prefetch_b8).
        __builtin_prefetch(&memory[(pa + STEP) * NCH + m], 0, 1);

        // D = A x B + C on the matrix pipe (f32, K=4).
        acc0 = __builtin_amdgcn_wmma_f32_16x16x4_f32(
            /*neg_a=*/false, a0, /*neg_b=*/false, b,
            /*c_mod=*/(short)0, acc0, /*reuse_a=*/false, /*reuse_b=*/false);
        acc1 = __builtin_amdgcn_wmma_f32_16x16x4_f32(
            /*neg_a=*/false, a1, /*neg_b=*/false, b,
            /*c_mod=*/(short)0, acc1, /*reuse_a=*/false, /*reuse_b=*/false);
    }

    // C/D layout: VGPR v, lanes 0-15 -> M=v, lanes 16-31 -> M=v+8; N = lane&15.
    #pragma unroll
    for (int v = 0; v < 8; ++v) {
        const int row = v + (half << 3);
        atomicAdd(&out[row * NCH + m],        acc0[v]);   // rays 0..15
        atomicAdd(&out[(row + 16) * NCH + m], acc1[v]);   // rays 16..31
    }
}

extern "C" void kernel_launch(void* const* d_in, const int* in_sizes, int n_in,
                              void* d_out, int out_size, void* d_ws, size_t ws_size,
                              hipStream_t stream) {
    const float* memory = (const float*)d_in[0];
    // d_in[1] is the meshgrid 'grid' array: coordinates are recomputed on the
    // fly from the linear index, so it is not read.
    const float* ray_o  = (const float*)d_in[2];
    const float* ray_d  = (const float*)d_in[3];
    float*       out    = (float*)d_out;
    (void)in_sizes; (void)n_in; (void)out_size; (void)d_ws; (void)ws_size;

    hpm_zero_out<<<(NRAYS * NCH + THREADS - 1) / THREADS, THREADS, 0, stream>>>(out);
    hpm_tube_read<<<BLOCKS, THREADS, 0, stream>>>(memory, ray_o, ray_d, out);
}